// GNN_11776800325917
// MI455X (gfx1250) — compile-verified
//
#include <hip/hip_runtime.h>
#include <hip/hip_bf16.h>

typedef __attribute__((ext_vector_type(2))) float v2f;
typedef __attribute__((ext_vector_type(8))) float v8f;

#define IN_DIM 384
#define HID    16
#define NCLS   2
#define BN_EPS 1e-5f

// ---- monotone float<->uint encoding so max() works as unsigned atomic ----
__device__ __forceinline__ unsigned encf(float f) {
    unsigned b = __float_as_uint(f);
    return (b & 0x80000000u) ? ~b : (b | 0x80000000u);
}
__device__ __forceinline__ float decf(unsigned u) {
    unsigned b = (u & 0x80000000u) ? (u & 0x7fffffffu) : ~u;
    return __uint_as_float(b);
}
#define ENC_NEG_INF 0x007FFFFFu   // encf(-inf)

// ------------------------------------------------------------------ fills
__global__ void fill_u32(unsigned* __restrict__ p, unsigned v, size_t n) {
    size_t i = (size_t)blockIdx.x * blockDim.x + threadIdx.x;
    if (i < n) p[i] = v;
}

// ------------------------------------------------------- GEMM1: x -> xpl,xpr
// xpl = x @ lin_l_w[:, :384].T   (mean-branch projection)
// xpr = x @ lin_r_w.T            (root projection)
// One wave per 16-row tile; two accumulators share the A operand.
__global__ void gemm_proj2(const float* __restrict__ x,
                           const float* __restrict__ wl,   // [16 x 768]
                           const float* __restrict__ wr,   // [16 x 384]
                           float* __restrict__ xpl, float* __restrict__ xpr,
                           int ntiles) {
    int wave = blockIdx.x * (blockDim.x >> 5) + (threadIdx.x >> 5);
    if (wave >= ntiles) return;                 // wave-uniform: EXEC stays all-1
    int lane = threadIdx.x & 31;
    int m = lane & 15, half = lane >> 4;
    int row0 = wave * 16;
    const float* arow = x  + (size_t)(row0 + m) * IN_DIM + half * 2;
    const float* brl  = wl + (size_t)m * (2 * IN_DIM) + half * 2;   // cols 0..383
    const float* brr  = wr + (size_t)m * IN_DIM + half * 2;
    v8f cl = {0.f,0.f,0.f,0.f,0.f,0.f,0.f,0.f};
    v8f cr = {0.f,0.f,0.f,0.f,0.f,0.f,0.f,0.f};
    for (int k = 0; k < IN_DIM; k += 4) {
        v2f a  = *(const v2f*)(arow + k);
        v2f bl = *(const v2f*)(brl + k);
        v2f br = *(const v2f*)(brr + k);
        cl = __builtin_amdgcn_wmma_f32_16x16x4_f32(false, a, false, bl, (short)0, cl, false, false);
        cr = __builtin_amdgcn_wmma_f32_16x16x4_f32(false, a, false, br, (short)0, cr, false, false);
    }
    for (int r = 0; r < 8; ++r) {
        int row = row0 + r + half * 8;
        xpl[(size_t)row * HID + m] = cl[r];
        xpr[(size_t)row * HID + m] = cr[r];
    }
}

// ------------------------------------------------- GEMM2: max_agg -> maxproj
// maxproj = max_agg @ lin_l_w[:, 384:768].T
__global__ void gemm_maxproj(const float* __restrict__ mx,  // [N x 384] decoded
                             const float* __restrict__ wl,  // [16 x 768]
                             float* __restrict__ out, int ntiles) {
    int wave = blockIdx.x * (blockDim.x >> 5) + (threadIdx.x >> 5);
    if (wave >= ntiles) return;
    int lane = threadIdx.x & 31;
    int m = lane & 15, half = lane >> 4;
    int row0 = wave * 16;
    const float* arow = mx + (size_t)(row0 + m) * IN_DIM + half * 2;
    const float* brow = wl + (size_t)m * (2 * IN_DIM) + IN_DIM + half * 2; // cols 384..767
    v8f c = {0.f,0.f,0.f,0.f,0.f,0.f,0.f,0.f};
    for (int k = 0; k < IN_DIM; k += 4) {
        v2f a = *(const v2f*)(arow + k);
        v2f b = *(const v2f*)(brow + k);
        c = __builtin_amdgcn_wmma_f32_16x16x4_f32(false, a, false, b, (short)0, c, false, false);
    }
    for (int r = 0; r < 8; ++r)
        out[(size_t)(row0 + r + half * 8) * HID + m] = c[r];
}

// ------------------------------------------------------------- edge kernel
// One 128-thread block per edge:
//   t<96   : 4 features each -> atomic max (u32-encoded) into maxbuf[i]
//   96..111: atomic add of projected source feature into sum_proj[i]
//   t==112 : degree count
__global__ void edge_aggregate(const int* __restrict__ ei,
                               const float* __restrict__ x,
                               const float* __restrict__ xpl,
                               float* __restrict__ sum_proj,
                               float* __restrict__ cnt,
                               unsigned* __restrict__ maxbuf, int E) {
    int e = blockIdx.x;
    if (e >= E) return;
    int t = threadIdx.x;
    int i = ei[e];          // target (aggregation destination)
    int j = ei[E + e];      // source (message)
    if (t < 96) {
        const float4 v = *(const float4*)(x + (size_t)j * IN_DIM + t * 4);
        unsigned* mp = maxbuf + (size_t)i * IN_DIM + t * 4;
        atomicMax(mp + 0, encf(v.x));
        atomicMax(mp + 1, encf(v.y));
        atomicMax(mp + 2, encf(v.z));
        atomicMax(mp + 3, encf(v.w));
    } else if (t < 96 + HID) {
        int k = t - 96;
        atomicAdd(sum_proj + (size_t)i * HID + k, xpl[(size_t)j * HID + k]);
    } else if (t == 96 + HID) {
        atomicAdd(cnt + i, 1.0f);
    }
}

// --------------------------------------------- decode maxbuf in place (u32->f32)
__global__ void decode_max(unsigned* __restrict__ buf, size_t total) {
    size_t idx = (size_t)blockIdx.x * blockDim.x + threadIdx.x;
    if (idx >= total) return;
    float f = decf(buf[idx]);
    if (!isfinite(f)) f = 0.0f;          // isolated nodes: PyG leaves 0
    ((float*)buf)[idx] = f;
}

// --------------------------------------- h = combine, accumulate BN statistics
__global__ void node_h_stats(const float* __restrict__ xpr,
                             const float* __restrict__ sum_proj,
                             const float* __restrict__ cnt,
                             const float* __restrict__ maxproj,
                             const float* __restrict__ bias,
                             float* __restrict__ h,
                             float* __restrict__ stats,   // [32] sums, sumsq
                             int N) {
    __shared__ float sred[2 * HID];
    if (threadIdx.x < 2 * HID) sred[threadIdx.x] = 0.0f;
    __syncthreads();
    int n = blockIdx.x * blockDim.x + threadIdx.x;
    if (n < N) {
        float invc = 1.0f / fmaxf(cnt[n], 1.0f);
        for (int k = 0; k < HID; ++k) {
            size_t o = (size_t)n * HID + k;
            float v = xpr[o] + sum_proj[o] * invc + maxproj[o] + bias[k];
            h[o] = v;
            atomicAdd(&sred[k], v);
            atomicAdd(&sred[HID + k], v * v);
        }
    }
    __syncthreads();
    if (threadIdx.x < 2 * HID) atomicAdd(&stats[threadIdx.x], sred[threadIdx.x]);
}

// ------------------------------------------------- BN scale/shift finalize
__global__ void bn_finalize(const float* __restrict__ stats,
                            const float* __restrict__ gamma,
                            const float* __restrict__ beta,
                            float* __restrict__ ss, int N) {
    int k = threadIdx.x;
    if (k < HID) {
        float invN = 1.0f / (float)N;
        float mu  = stats[k] * invN;
        float var = fmaxf(stats[HID + k] * invN - mu * mu, 0.0f);  // biased var
        float scale = gamma[k] * rsqrtf(var + BN_EPS);
        ss[k]       = scale;
        ss[HID + k] = beta[k] - mu * scale;
    }
}

// ------------------------------------------- normalize + ReLU + mean-pool acc
__global__ void norm_relu_pool(const float* __restrict__ h,
                               const float* __restrict__ ss,
                               const int* __restrict__ batch,
                               float* __restrict__ pooled,
                               float* __restrict__ gcnt, int N) {
    int n = blockIdx.x * blockDim.x + threadIdx.x;
    if (n >= N) return;
    int b = batch[n];
    for (int k = 0; k < HID; ++k) {
        float v = fmaxf(h[(size_t)n * HID + k] * ss[k] + ss[HID + k], 0.0f);
        atomicAdd(pooled + (size_t)b * HID + k, v);
    }
    atomicAdd(gcnt + b, 1.0f);
}

// ------------------------------------------------------------- final FC head
__global__ void final_fc(const float* __restrict__ pooled,
                         const float* __restrict__ gcnt,
                         const float* __restrict__ fcw,  // [2 x 16]
                         const float* __restrict__ fcb,
                         float* __restrict__ out, int G) {
    int g = threadIdx.x;
    if (g >= G) return;
    float invg = 1.0f / fmaxf(gcnt[g], 1.0f);
    for (int c = 0; c < NCLS; ++c) {
        float acc = fcb[c];
        for (int k = 0; k < HID; ++k)
            acc += pooled[(size_t)g * HID + k] * invg * fcw[c * HID + k];
        out[g * NCLS + c] = acc;
    }
}

extern "C" void kernel_launch(void* const* d_in, const int* in_sizes, int n_in,
                              void* d_out, int out_size, void* d_ws, size_t ws_size,
                              hipStream_t stream) {
    const float* x     = (const float*)d_in[0];
    const int*   ei    = (const int*)  d_in[1];
    const int*   batch = (const int*)  d_in[2];
    const float* llw   = (const float*)d_in[4];   // [16 x 768]
    const float* llb   = (const float*)d_in[5];   // [16]
    const float* lrw   = (const float*)d_in[6];   // [16 x 384]
    const float* gam   = (const float*)d_in[7];
    const float* bet   = (const float*)d_in[8];
    const float* fcw   = (const float*)d_in[9];   // [2 x 16]
    const float* fcb   = (const float*)d_in[10];
    float* out = (float*)d_out;

    const int N = in_sizes[0] / IN_DIM;       // 50000
    const int E = in_sizes[1] / 2;            // 200000
    const int G = out_size / NCLS;            // 64
    const int ntiles = (N + 15) / 16;         // 3125 (N % 16 == 0)

    // ---- workspace layout (floats) ----
    float* ws = (float*)d_ws;
    size_t n16 = (size_t)N * HID;
    float*    sum_proj = ws;                          // N*16
    float*    cnt      = sum_proj + n16;              // N
    float*    xpl      = cnt + N;                     // N*16
    float*    xpr      = xpl + n16;                   // N*16
    unsigned* maxbuf   = (unsigned*)(xpr + n16);      // N*384 (u32 then f32)
    float*    maxproj  = (float*)maxbuf + (size_t)N * IN_DIM;  // N*16
    float*    h        = maxproj + n16;               // N*16
    float*    stats    = h + n16;                     // 32
    float*    ss       = stats + 32;                  // 32
    float*    pooled   = ss + 32;                     // G*16
    float*    gcnt     = pooled + (size_t)G * HID;    // G
    (void)ws_size; (void)n_in;

    // ---- init accumulators ----
    {
        size_t nz = n16 + (size_t)N;                  // sum_proj + cnt (contiguous)
        fill_u32<<<(unsigned)((nz + 255) / 256), 256, 0, stream>>>((unsigned*)sum_proj, 0u, nz);
        size_t nm = (size_t)N * IN_DIM;
        fill_u32<<<(unsigned)((nm + 255) / 256), 256, 0, stream>>>(maxbuf, ENC_NEG_INF, nm);
        size_t nmisc = 32 + 32 + (size_t)G * HID + G; // stats, ss, pooled, gcnt
        fill_u32<<<(unsigned)((nmisc + 255) / 256), 256, 0, stream>>>((unsigned*)stats, 0u, nmisc);
    }

    // ---- projections of x (WMMA fp32) ----
    gemm_proj2<<<(ntiles + 7) / 8, 256, 0, stream>>>(x, llw, lrw, xpl, xpr, ntiles);

    // ---- edge aggregation: native u32 atomic max + projected atomic sums ----
    edge_aggregate<<<E, 128, 0, stream>>>(ei, x, xpl, sum_proj, cnt, maxbuf, E);

    // ---- decode max buffer in place ----
    {
        size_t total = (size_t)N * IN_DIM;
        decode_max<<<(unsigned)((total + 255) / 256), 256, 0, stream>>>(maxbuf, total);
    }

    // ---- project max aggregation (WMMA fp32) ----
    gemm_maxproj<<<(ntiles + 7) / 8, 256, 0, stream>>>((const float*)maxbuf, llw, maxproj, ntiles);

    // ---- combine + BN statistics ----
    node_h_stats<<<(N + 255) / 256, 256, 0, stream>>>(xpr, sum_proj, cnt, maxproj, llb, h, stats, N);

    // ---- BN finalize ----
    bn_finalize<<<1, 32, 0, stream>>>(stats, gam, bet, ss, N);

    // ---- normalize + ReLU + pooling ----
    norm_relu_pool<<<(N + 255) / 256, 256, 0, stream>>>(h, ss, batch, pooled, gcnt, N);

    // ---- final FC ----
    final_fc<<<1, 64, 0, stream>>>(pooled, gcnt, fcw, fcb, out, G);
}